// GINEncoder_44461501448893
// MI455X (gfx1250) — compile-verified
//
#include <hip/hip_runtime.h>
#include <hip/hip_bf16.h>

typedef __attribute__((ext_vector_type(2))) float v2f;
typedef __attribute__((ext_vector_type(8))) float v8f;

#define NNODES 50000
#define NEDGES 1000000
#define HID 64
#define NGRAPH 512
#define ULD 66  // padded LDS row stride (floats) to avoid bank conflicts

// ---------------- WMMA helpers (fp32 16x16x4) ----------------

__device__ __forceinline__ v8f wmma4(v2f a, v2f b, v8f c) {
  // D = A(16x4) * B(4x16) + C, fp32, wave32
  return __builtin_amdgcn_wmma_f32_16x16x4_f32(false, a, false, b, (short)0, c, false, false);
}

__device__ __forceinline__ void zero4(v8f* a) {
  v8f z = {0.f, 0.f, 0.f, 0.f, 0.f, 0.f, 0.f, 0.f};
#pragma unroll
  for (int j = 0; j < 4; ++j) a[j] = z;
}

// Pre-swizzle B[K][64] (row-major) into per-lane fragment order so each
// WMMA B operand is a single contiguous 8-byte LDS load:
//   frag index i = (kk*4 + j)*32 + lane ; lane: n=lane&15, kb=(lane>>4)*2
//   bswz[2i+p] = B[4*kk + kb + p][j*16 + n]
__device__ __forceinline__ void fill_bswz(float* bswz, const float* __restrict__ B, int K) {
  const int total = K * 32;
  for (int i = threadIdx.x; i < total; i += blockDim.x) {
    const int lane = i & 31;
    const int j = (i >> 5) & 3;
    const int kk = i >> 7;
    const int n = lane & 15;
    const int kb = (lane >> 4) << 1;
    const int kr = (kk << 2) + kb;
    bswz[2 * i]     = B[kr * HID + j * 16 + n];
    bswz[2 * i + 1] = B[(kr + 1) * HID + j * 16 + n];
  }
}

// One 16-row x 64-col tile GEMM: acc[4] (N-tiles of 16) += A_tile(16xK) * B(Kx64).
// If SUM, A element = A[..] + A2[..] (used for z0 = h + agg).
template<int K4, bool SUM>
__device__ __forceinline__ void gemm_tile(const float* A, const float* A2, int ldA,
                                          const float* bswz, v8f* acc, int lane) {
  const int m  = lane & 15;
  const int kb = (lane >> 4) << 1;
  const float* ar  = A + m * ldA + kb;
  const float* ar2 = SUM ? (A2 + m * ldA + kb) : nullptr;
  const float* bp  = bswz + lane * 2;
#pragma unroll
  for (int kk = 0; kk < K4; ++kk) {
    v2f a = *(const v2f*)(ar + kk * 4);
    if (SUM) { v2f a2 = *(const v2f*)(ar2 + kk * 4); a = a + a2; }
#pragma unroll
    for (int j = 0; j < 4; ++j) {
      v2f b = *(const v2f*)(bp + kk * 256 + j * 64);
      acc[j] = wmma4(a, b, acc[j]);
    }
  }
}

__device__ __forceinline__ float atomAddF(float* p, float v) {
  return __hip_atomic_fetch_add(p, v, __ATOMIC_RELAXED, __HIP_MEMORY_SCOPE_AGENT);
}

// ---------------- kernels ----------------

// Wcomb = W2e @ Wl ; bcomb = b2e @ Wl + bl   (tiny, one launch per layer)
__global__ void k_prep(const float* __restrict__ W2e, const float* __restrict__ b2e,
                       const float* __restrict__ Wl, const float* __restrict__ bl,
                       float* __restrict__ Wc, float* __restrict__ bc) {
  int i = blockIdx.x * blockDim.x + threadIdx.x;
  if (i < 64 * 64) {
    int r = i >> 6, c = i & 63;
    float s = 0.f;
    for (int k = 0; k < 64; ++k) s += W2e[r * 64 + k] * Wl[k * 64 + c];
    Wc[i] = s;
  }
  if (i < 64) {
    float s = bl[i];
    for (int k = 0; k < 64; ++k) s += b2e[k] * Wl[k * 64 + i];
    bc[i] = s;
  }
}

// h = x @ Wn + bn    ([50000,128] @ [128,64])
__global__ void __launch_bounds__(256) k_node_emb(
    const float* __restrict__ x, const float* __restrict__ W,
    const float* __restrict__ b, float* __restrict__ h, int ntiles) {
  __shared__ __align__(16) float bswz[128 * HID];
  fill_bswz(bswz, W, 128);
  __syncthreads();
  const int lane = threadIdx.x & 31;
  const int nn = lane & 15, hi = lane >> 4;
  const int wid = (blockIdx.x * blockDim.x + threadIdx.x) >> 5;
  const int nw  = (gridDim.x * blockDim.x) >> 5;
  float bj[4];
#pragma unroll
  for (int j = 0; j < 4; ++j) bj[j] = b[j * 16 + nn];
  for (int t = wid; t < ntiles; t += nw) {
    v8f acc[4]; zero4(acc);
    gemm_tile<32, false>(x + t * 16 * 128, nullptr, 128, bswz, acc, lane);
#pragma unroll
    for (int r = 0; r < 8; ++r) {
      float* out = h + (t * 16 + r + 8 * hi) * HID + nn;
#pragma unroll
      for (int j = 0; j < 4; ++j) out[j * 16] = acc[j][r] + bj[j];
    }
  }
}

// Fused per-layer edge pipeline:
//   u  = relu(edge_attr @ W1e + b1e)        (WMMA, K=32)
//   ee = u @ Wcomb + bcomb                  (WMMA, K=64, A staged via LDS)
//   msg = relu(h[src] + ee); agg[dst] += msg (L2 gather + fp32 global atomics)
__global__ void __launch_bounds__(128) k_edge_msg(
    const float* __restrict__ ea, const float* __restrict__ W1, const float* __restrict__ b1,
    const float* __restrict__ Wc, const float* __restrict__ bc, const float* __restrict__ h,
    const int* __restrict__ src, const int* __restrict__ dst,
    float* __restrict__ agg, int ntiles) {
  __shared__ __align__(16) float bswz1[32 * HID];
  __shared__ __align__(16) float bswzc[64 * HID];
  __shared__ __align__(16) float ust[4][16 * ULD];
  fill_bswz(bswz1, W1, 32);
  fill_bswz(bswzc, Wc, 64);
  __syncthreads();
  const int lane = threadIdx.x & 31;
  const int warp = threadIdx.x >> 5;
  const int nn = lane & 15, hi = lane >> 4;
  float* uL = ust[warp];
  const int wid = (blockIdx.x * blockDim.x + threadIdx.x) >> 5;
  const int nw  = (gridDim.x * blockDim.x) >> 5;
  float b1j[4], bcj[4];
#pragma unroll
  for (int j = 0; j < 4; ++j) { b1j[j] = b1[j * 16 + nn]; bcj[j] = bc[j * 16 + nn]; }
  for (int t = wid; t < ntiles; t += nw) {
    __builtin_prefetch(ea + (t + nw) * 512 + lane * 16, 0, 1);
    v8f acc1[4]; zero4(acc1);
    gemm_tile<8, false>(ea + t * 16 * 32, nullptr, 32, bswz1, acc1, lane);
#pragma unroll
    for (int r = 0; r < 8; ++r)
#pragma unroll
      for (int j = 0; j < 4; ++j)
        uL[(r + 8 * hi) * ULD + j * 16 + nn] = fmaxf(acc1[j][r] + b1j[j], 0.f);
    __asm__ volatile("" ::: "memory");  // LDS in-order per wave; block compiler reordering
    v8f acc2[4]; zero4(acc2);
    gemm_tile<16, false>(uL, nullptr, ULD, bswzc, acc2, lane);
#pragma unroll
    for (int r = 0; r < 8; ++r) {
      const int eid = t * 16 + r + 8 * hi;
      const int s = src[eid];
      const int d = dst[eid];
      const float* hr = h + s * HID + nn;
      float* ar = agg + d * HID + nn;
#pragma unroll
      for (int j = 0; j < 4; ++j) {
        float v = fmaxf(acc2[j][r] + bcj[j] + hr[j * 16], 0.f);
        atomAddF(ar + j * 16, v);
      }
    }
  }
}

// h2 = relu((h+agg) @ W1 + b1) @ W2 + b2 ; accumulate BN sum/sumsq in regs.
__global__ void __launch_bounds__(128) k_node_mlp(
    const float* __restrict__ h, const float* __restrict__ agg,
    const float* __restrict__ W1, const float* __restrict__ b1,
    const float* __restrict__ W2, const float* __restrict__ b2,
    float* __restrict__ h2, float* __restrict__ bnsum, float* __restrict__ bnsq,
    int ntiles) {
  __shared__ __align__(16) float bswz1[64 * HID];
  __shared__ __align__(16) float bswz2[64 * HID];
  __shared__ __align__(16) float ust[4][16 * ULD];
  fill_bswz(bswz1, W1, 64);
  fill_bswz(bswz2, W2, 64);
  __syncthreads();
  const int lane = threadIdx.x & 31;
  const int warp = threadIdx.x >> 5;
  const int nn = lane & 15, hi = lane >> 4;
  float* uL = ust[warp];
  const int wid = (blockIdx.x * blockDim.x + threadIdx.x) >> 5;
  const int nw  = (gridDim.x * blockDim.x) >> 5;
  float b1j[4], b2j[4];
#pragma unroll
  for (int j = 0; j < 4; ++j) { b1j[j] = b1[j * 16 + nn]; b2j[j] = b2[j * 16 + nn]; }
  float sreg[4] = {0.f, 0.f, 0.f, 0.f};
  float qreg[4] = {0.f, 0.f, 0.f, 0.f};
  for (int t = wid; t < ntiles; t += nw) {
    v8f acc1[4]; zero4(acc1);
    gemm_tile<16, true>(h + t * 1024, agg + t * 1024, 64, bswz1, acc1, lane);
#pragma unroll
    for (int r = 0; r < 8; ++r)
#pragma unroll
      for (int j = 0; j < 4; ++j)
        uL[(r + 8 * hi) * ULD + j * 16 + nn] = fmaxf(acc1[j][r] + b1j[j], 0.f);
    __asm__ volatile("" ::: "memory");
    v8f acc2[4]; zero4(acc2);
    gemm_tile<16, false>(uL, nullptr, ULD, bswz2, acc2, lane);
#pragma unroll
    for (int r = 0; r < 8; ++r) {
      float* out = h2 + (t * 16 + r + 8 * hi) * HID + nn;
#pragma unroll
      for (int j = 0; j < 4; ++j) {
        float v = acc2[j][r] + b2j[j];
        out[j * 16] = v;
        sreg[j] += v;
        qreg[j] += v * v;
      }
    }
  }
#pragma unroll
  for (int j = 0; j < 4; ++j) {
    atomAddF(bnsum + j * 16 + nn, sreg[j]);
    atomAddF(bnsq  + j * 16 + nn, qreg[j]);
  }
}

__global__ void k_bn_fin(const float* __restrict__ sum, const float* __restrict__ sq,
                         const float* __restrict__ gamma, const float* __restrict__ beta,
                         float* __restrict__ scale, float* __restrict__ shift) {
  int c = threadIdx.x;
  if (c < HID) {
    const float inv_n = 1.0f / (float)NNODES;
    float mu  = sum[c] * inv_n;
    float var = sq[c] * inv_n - mu * mu;           // biased variance (matches ref)
    float s = gamma[c] * rsqrtf(var + 1e-5f);
    scale[c] = s;
    shift[c] = beta[c] - mu * s;
  }
}

__global__ void k_bn_apply(const float* __restrict__ h2, const float* __restrict__ sc,
                           const float* __restrict__ sh, float* __restrict__ h) {
  int i = blockIdx.x * blockDim.x + threadIdx.x;  // exactly NNODES*64 threads
  int c = i & 63;
  h[i] = fmaxf(h2[i] * sc[c] + sh[c], 0.f);
}

__global__ void k_pool(const float* __restrict__ h, const int* __restrict__ batch,
                       float* __restrict__ gadd, float* __restrict__ cnts) {
  int i = blockIdx.x * blockDim.x + threadIdx.x;  // exactly NNODES*64 threads
  int n = i >> 6, c = i & 63;
  int b = batch[n];
  atomAddF(gadd + b * HID + c, h[i]);
  if (c == 0) atomAddF(cnts + b, 1.0f);
}

__global__ void k_pool_fin(const float* __restrict__ gadd, const float* __restrict__ cnts,
                           float* __restrict__ out) {
  int i = blockIdx.x * blockDim.x + threadIdx.x;  // 512*128 threads
  int gi = i >> 7, c = i & 127;
  float v;
  if (c < HID) v = gadd[gi * HID + c] / fmaxf(cnts[gi], 1.0f);
  else         v = gadd[gi * HID + (c - HID)];
  out[i] = v;
}

// z = relu(g @ Wp1 + bp1) @ Wp2 + bp2   ([512,128] -> [512,64])
__global__ void __launch_bounds__(64) k_proj(
    const float* __restrict__ g, const float* __restrict__ W1, const float* __restrict__ b1,
    const float* __restrict__ W2, const float* __restrict__ b2, float* __restrict__ z) {
  __shared__ __align__(16) float bswz1[128 * HID];
  __shared__ __align__(16) float bswz2[64 * HID];
  __shared__ __align__(16) float ust[2][16 * ULD];
  fill_bswz(bswz1, W1, 128);
  fill_bswz(bswz2, W2, 64);
  __syncthreads();
  const int lane = threadIdx.x & 31;
  const int warp = threadIdx.x >> 5;
  const int nn = lane & 15, hi = lane >> 4;
  float* uL = ust[warp];
  const int wid = (blockIdx.x * blockDim.x + threadIdx.x) >> 5;
  const int nw  = (gridDim.x * blockDim.x) >> 5;
  float b1j[4], b2j[4];
#pragma unroll
  for (int j = 0; j < 4; ++j) { b1j[j] = b1[j * 16 + nn]; b2j[j] = b2[j * 16 + nn]; }
  for (int t = wid; t < NGRAPH / 16; t += nw) {
    v8f acc1[4]; zero4(acc1);
    gemm_tile<32, false>(g + t * 16 * 128, nullptr, 128, bswz1, acc1, lane);
#pragma unroll
    for (int r = 0; r < 8; ++r)
#pragma unroll
      for (int j = 0; j < 4; ++j)
        uL[(r + 8 * hi) * ULD + j * 16 + nn] = fmaxf(acc1[j][r] + b1j[j], 0.f);
    __asm__ volatile("" ::: "memory");
    v8f acc2[4]; zero4(acc2);
    gemm_tile<16, false>(uL, nullptr, ULD, bswz2, acc2, lane);
#pragma unroll
    for (int r = 0; r < 8; ++r) {
      float* out = z + (t * 16 + r + 8 * hi) * HID + nn;
#pragma unroll
      for (int j = 0; j < 4; ++j) out[j * 16] = acc2[j][r] + b2j[j];
    }
  }
}

// ---------------- host orchestration ----------------

extern "C" void kernel_launch(void* const* d_in, const int* in_sizes, int n_in,
                              void* d_out, int out_size, void* d_ws, size_t ws_size,
                              hipStream_t stream) {
  (void)n_in; (void)out_size; (void)ws_size;
  const float *x, *ea, *Wn, *bn_, *We1, *be1, *We2, *be2, *Wp1, *bp1, *Wp2, *bp2;
  const float *Wl[3], *bl[3], *Wm1[3], *bm1[3], *Wm2[3], *bm2[3], *gam[3], *bet[3];
  const int *ei, *batch;

  // Flattening-order detection: insertion order -> d_in[0] is x (6.4M elems);
  // jax sorted-key order -> d_in[0] is batch (50000 elems).
  if (in_sizes[0] == NNODES) {  // sorted-key order
    batch = (const int*)d_in[0];
    ea = (const float*)d_in[1];
    ei = (const int*)d_in[2];
    We1 = (const float*)d_in[3]; be1 = (const float*)d_in[4];
    We2 = (const float*)d_in[5]; be2 = (const float*)d_in[6];
    for (int l = 0; l < 3; ++l) {
      int p = 7 + 8 * l;
      gam[l] = (const float*)d_in[p + 0]; bet[l] = (const float*)d_in[p + 1];
      Wl[l]  = (const float*)d_in[p + 2]; bl[l]  = (const float*)d_in[p + 3];
      Wm1[l] = (const float*)d_in[p + 4]; bm1[l] = (const float*)d_in[p + 5];
      Wm2[l] = (const float*)d_in[p + 6]; bm2[l] = (const float*)d_in[p + 7];
    }
    Wn = (const float*)d_in[31]; bn_ = (const float*)d_in[32];
    Wp1 = (const float*)d_in[33]; bp1 = (const float*)d_in[34];
    Wp2 = (const float*)d_in[35]; bp2 = (const float*)d_in[36];
    x = (const float*)d_in[37];
  } else {  // insertion order
    x = (const float*)d_in[0];
    ea = (const float*)d_in[1];
    ei = (const int*)d_in[2];
    batch = (const int*)d_in[3];
    Wn = (const float*)d_in[4]; bn_ = (const float*)d_in[5];
    We1 = (const float*)d_in[6]; be1 = (const float*)d_in[7];
    We2 = (const float*)d_in[8]; be2 = (const float*)d_in[9];
    for (int l = 0; l < 3; ++l) {
      int p = 10 + 8 * l;
      Wl[l]  = (const float*)d_in[p + 0]; bl[l]  = (const float*)d_in[p + 1];
      Wm1[l] = (const float*)d_in[p + 2]; bm1[l] = (const float*)d_in[p + 3];
      Wm2[l] = (const float*)d_in[p + 4]; bm2[l] = (const float*)d_in[p + 5];
      gam[l] = (const float*)d_in[p + 6]; bet[l] = (const float*)d_in[p + 7];
    }
    Wp1 = (const float*)d_in[34]; bp1 = (const float*)d_in[35];
    Wp2 = (const float*)d_in[36]; bp2 = (const float*)d_in[37];
  }

  // Workspace carve (fp32): ~38.6 MB total
  float* w = (float*)d_ws;
  float* h    = w; w += NNODES * HID;       // 3.2M
  float* agg  = w; w += NNODES * HID;       // 3.2M
  float* h2   = w; w += NNODES * HID;       // 3.2M
  float* Wc   = w; w += 3 * 64 * 64;
  float* bc   = w; w += 3 * 64;
  float* bnst = w; w += 128;                // sum[64] | sumsq[64]
  float* bnsc = w; w += 128;                // scale[64] | shift[64]
  float* gadd = w; w += NGRAPH * HID;       // followed immediately by cnts
  float* cnts = w; w += NGRAPH;

  const int src_off = 0, dst_off = NEDGES;

  // Combined edge weights per layer (folds edge_enc second matmul into layer lin)
  for (int l = 0; l < 3; ++l)
    k_prep<<<16, 256, 0, stream>>>(We2, be2, Wl[l], bl[l], Wc + l * 4096, bc + l * 64);

  // Node embedding
  k_node_emb<<<128, 256, 0, stream>>>(x, Wn, bn_, h, NNODES / 16);

  for (int l = 0; l < 3; ++l) {
    hipMemsetAsync(agg, 0, (size_t)NNODES * HID * sizeof(float), stream);
    hipMemsetAsync(bnst, 0, 128 * sizeof(float), stream);
    k_edge_msg<<<1024, 128, 0, stream>>>(ea, We1, be1, Wc + l * 4096, bc + l * 64,
                                         h, ei + src_off, ei + dst_off, agg, NEDGES / 16);
    k_node_mlp<<<256, 128, 0, stream>>>(h, agg, Wm1[l], bm1[l], Wm2[l], bm2[l],
                                        h2, bnst, bnst + 64, NNODES / 16);
    k_bn_fin<<<1, 64, 0, stream>>>(bnst, bnst + 64, gam[l], bet[l], bnsc, bnsc + 64);
    k_bn_apply<<<12500, 256, 0, stream>>>(h2, bnsc, bnsc + 64, h);
  }

  // Pooling: g = [mean | sum] written straight into d_out[0 : 512*128]
  hipMemsetAsync(gadd, 0, (size_t)(NGRAPH * HID + NGRAPH) * sizeof(float), stream);
  k_pool<<<12500, 256, 0, stream>>>(h, batch, gadd, cnts);
  k_pool_fin<<<256, 256, 0, stream>>>(gadd, cnts, (float*)d_out);

  // Projection head: z into d_out[512*128 : ]
  k_proj<<<16, 64, 0, stream>>>((const float*)d_out, Wp1, bp1, Wp2, bp2,
                                (float*)d_out + NGRAPH * 2 * HID);
}